// GraphFilter_66554813219094
// MI455X (gfx1250) — compile-verified
//
#include <hip/hip_runtime.h>
#include <hip/hip_bf16.h>

// Problem constants (match reference)
#define BB 8
#define NN 2048
#define FF 8
#define KK 5

// Tiling for the S^T * x WMMA pass
#define MCH   64              // m-chunk staged in LDS per iteration
#define SROW  68              // padded LDS row stride (floats) -> conflict-free ds_load_b64
#define NWAVE 4               // waves per workgroup (one 16-row n-tile each)

typedef __attribute__((ext_vector_type(2))) float v2f;
typedef __attribute__((ext_vector_type(4))) float v4f;
typedef __attribute__((ext_vector_type(8))) float v8f;

// ---------------------------------------------------------------------------
// Pass kernel: xout[b, n, f] = sum_m S[b, m, n] * xin[b, m, f]   (x <- S^T x)
// One wave computes a 16(n) x 16(f) WMMA tile (cols 8..15 discarded), reducing
// over m = 0..2047 with V_WMMA_F32_16X16X4_F32. Software-pipelined: chunk i+1
// streams from global into registers while chunk i computes out of LDS.
// ---------------------------------------------------------------------------
__global__ __launch_bounds__(32 * NWAVE)
void spmv_wmma_kernel(const float* __restrict__ S,
                      const float* __restrict__ xin,
                      float* __restrict__ xout) {
  // S block, transposed per wave: s_lds[wave][n(16)][m(64)] (padded rows)
  __shared__ float s_lds[NWAVE * 16 * SROW];
  // x chunk, transposed, shared by all waves: x_lds[f(16, 8 valid)][m(64)]
  __shared__ float x_lds[16 * SROW];

  const int b    = blockIdx.y;
  const int tid  = threadIdx.x;
  const int wave = tid >> 5;
  const int lane = tid & 31;
  const int n0   = blockIdx.x * (16 * NWAVE) + wave * 16;

  const float* Sb = S   + (size_t)b * NN * NN;
  const float* xb = xin + (size_t)b * NN * FF;

  const int l15 = lane & 15;   // A: M(=n) index / B: N(=f) index / D: column
  const int lhi = lane >> 4;   // selects K pair {0,1} vs {2,3} within a step

  float* s_wave = s_lds + wave * 16 * SROW;
  // per-lane fragment base (float index); += 4*step inside the chunk
  const int frag_base = l15 * SROW + lhi * 2;

  // cooperative-load index precomputation
  const int xm   = tid >> 1;           // 0..63   (x chunk row)
  const int xf0  = (tid & 1) * 4;      // 0 or 4  (x chunk col group)
  const int mrow = lane >> 2;          // 0..7    (S block row within round)
  const int f4   = (lane & 3) * 4;     // 0,4,8,12 (S block col group)

  v8f acc = {};                // 16x16 f32 accumulator, lives across all chunks

  // --- pipeline prologue: chunk 0 -> registers -------------------------------
  v4f x_pref;
  v4f s_pref[8];
  x_pref = *(const v4f*)(xb + (size_t)xm * FF + xf0);
#pragma unroll
  for (int r = 0; r < 8; ++r)
    s_pref[r] = *(const v4f*)(Sb + (size_t)(mrow + r * 8) * NN + n0 + f4);

  for (int mc = 0; mc < NN; mc += MCH) {
    __syncthreads();           // previous chunk's compute done before overwrite

    // --- spill prefetched registers into transposed LDS layouts ---
    x_lds[(xf0 + 0) * SROW + xm] = x_pref.x;
    x_lds[(xf0 + 1) * SROW + xm] = x_pref.y;
    x_lds[(xf0 + 2) * SROW + xm] = x_pref.z;
    x_lds[(xf0 + 3) * SROW + xm] = x_pref.w;
#pragma unroll
    for (int r = 0; r < 8; ++r) {
      const int m = mrow + r * 8;
      s_wave[(f4 + 0) * SROW + m] = s_pref[r].x;
      s_wave[(f4 + 1) * SROW + m] = s_pref[r].y;
      s_wave[(f4 + 2) * SROW + m] = s_pref[r].z;
      s_wave[(f4 + 3) * SROW + m] = s_pref[r].w;
    }

    // --- issue next chunk's global loads (overlap with the WMMA stream) ---
    {
      int mc_next = mc + MCH;
      if (mc_next >= NN) mc_next = 0;            // harmless wrap, uniform
      x_pref = *(const v4f*)(xb + (size_t)(mc_next + xm) * FF + xf0);
#pragma unroll
      for (int r = 0; r < 8; ++r)
        s_pref[r] =
            *(const v4f*)(Sb + (size_t)(mc_next + mrow + r * 8) * NN + n0 + f4);
    }

    __syncthreads();           // LDS staging visible to all waves

    // --- 16 WMMA steps: K=4 each -> K=64 per chunk ---
#pragma unroll
    for (int s = 0; s < 16; ++s) {
      v2f a  = *(const v2f*)(s_wave + frag_base + 4 * s);  // ds_load_b64
      v2f bf = *(const v2f*)(x_lds  + frag_base + 4 * s);  // ds_load_b64
      acc = __builtin_amdgcn_wmma_f32_16x16x4_f32(
          /*neg_a=*/false, a, /*neg_b=*/false, bf,
          /*c_mod=*/(short)0, acc, /*reuse_a=*/false, /*reuse_b=*/false);
    }
  }

  // D layout: VGPR r, lane L -> (M = r + (L>>4)*8, N = L&15). M = n row, N = f.
  if (l15 < FF) {
    float* yb = xout + (size_t)b * NN * FF;
#pragma unroll
    for (int r = 0; r < 8; ++r) {
      const int n = n0 + r + lhi * 8;
      yb[(size_t)n * FF + l15] = acc[r];
    }
  }
}

// ---------------------------------------------------------------------------
// Epilogue: y[b,n,:] (+)= xk[b,n,:] @ Wk  (+ bias when initializing)
// One thread per (b,n) row; 8x8 MAC, negligible traffic (~1.5 MB total/pass).
// ---------------------------------------------------------------------------
__global__ __launch_bounds__(256)
void yw_kernel(const float* __restrict__ xk, const float* __restrict__ Wk,
               const float* __restrict__ bias, float* __restrict__ y,
               int accumulate) {
  const int i = blockIdx.x * blockDim.x + threadIdx.x;   // row over B*N
  if (i >= BB * NN) return;
  const float* xr = xk + (size_t)i * FF;
  float xv[FF];
#pragma unroll
  for (int f = 0; f < FF; ++f) xv[f] = xr[f];
#pragma unroll
  for (int g = 0; g < FF; ++g) {
    float s = accumulate ? y[(size_t)i * FF + g] : bias[g];
#pragma unroll
    for (int f = 0; f < FF; ++f) s += xv[f] * Wk[f * FF + g];
    y[(size_t)i * FF + g] = s;
  }
}

// ---------------------------------------------------------------------------
extern "C" void kernel_launch(void* const* d_in, const int* in_sizes, int n_in,
                              void* d_out, int out_size, void* d_ws, size_t ws_size,
                              hipStream_t stream) {
  const float* x    = (const float*)d_in[0];   // [B, N, F]
  const float* S    = (const float*)d_in[1];   // [B, N, N]
  const float* W    = (const float*)d_in[2];   // [K, F, F]
  const float* bias = (const float*)d_in[3];   // [F]
  float* y = (float*)d_out;                    // [B, N, F]

  float* buf0 = (float*)d_ws;                  // ping-pong x_k buffers
  float* buf1 = buf0 + (size_t)BB * NN * FF;   // needs 2 * 512 KB of d_ws

  const dim3 gw(NN / (16 * NWAVE), BB);        // (32, 8) workgroups of 128
  const int rows = BB * NN;
  const int nb = (rows + 255) / 256;

  // k = 0: y = x @ W0 + bias
  yw_kernel<<<nb, 256, 0, stream>>>(x, W, bias, y, 0);

  const float* src = x;
  float* dst = buf0;
  for (int k = 1; k < KK; ++k) {
    spmv_wmma_kernel<<<gw, 32 * NWAVE, 0, stream>>>(S, src, dst);
    yw_kernel<<<nb, 256, 0, stream>>>(dst, W + (size_t)k * FF * FF, bias, y, 1);
    src = dst;
    dst = (dst == buf0) ? buf1 : buf0;
  }
}